// GCN_25177098289682
// MI455X (gfx1250) — compile-verified
//
#include <hip/hip_runtime.h>
#include <math.h>

// ---------------------------------------------------------------------------
// GCN (GCNConv + BN + ReLU) x2 + GCNConv + log_softmax for MI455X (gfx1250).
// fp32 end-to-end; dense projections use V_WMMA_F32_16X16X4_F32 with the
// B-panel staged in LDS (constant-offset ds reads) and K=128 fully unrolled.
// ---------------------------------------------------------------------------

typedef float v2f __attribute__((ext_vector_type(2)));
typedef float v8f __attribute__((ext_vector_type(8)));

#define BN_EPS 1e-5f
#define GK 128          // inner dimension of every projection in this model

// ------------------------------- utility -----------------------------------
__global__ void __launch_bounds__(256) k_zero(float* __restrict__ p, int n) {
  int i = blockIdx.x * blockDim.x + threadIdx.x;
  if (i < n) p[i] = 0.0f;
}

// deg[dst] += 1 for every real edge (self-loop folded into k_dinv as +1)
__global__ void __launch_bounds__(256)
k_degree(const int* __restrict__ dst, float* __restrict__ deg, int E) {
  int e = blockIdx.x * blockDim.x + threadIdx.x;
  if (e < E) atomicAdd(&deg[dst[e]], 1.0f);
}

__global__ void __launch_bounds__(256)
k_dinv(const float* __restrict__ deg, float* __restrict__ dinv, int n) {
  int i = blockIdx.x * blockDim.x + threadIdx.x;
  if (i < n) dinv[i] = rsqrtf(deg[i] + 1.0f);  // +1 = self loop; deg >= 1 always
}

// ------------------------- WMMA fp32 GEMM ----------------------------------
// h[M,C] = A[M,GK] @ W[GK,C]; fused epilogue also writes
// agg[M,C] = h * dinv(row)^2 + bias[col]   (self-loop + bias fold).
// Block: 256 threads = 8 waves = 8 row tiles; one 16-column tile per block
// (blockIdx.y).  B panel (GK x 16, 8 KB) staged in LDS once per block; the
// unrolled K loop is then global_load_b64(A) + 2x ds_load_b32(B) + v_wmma,
// all with immediate offsets.  M must be a multiple of 16. C may be ragged.
__global__ void __launch_bounds__(256)
k_gemm_wmma(const float* __restrict__ A, const float* __restrict__ W,
            const float* __restrict__ dinv, const float* __restrict__ bias,
            float* __restrict__ h, float* __restrict__ agg,
            int M, int C) {
  __shared__ float bp[GK * 16];                 // [k][col] panel, 8 KB
  const int col0 = blockIdx.y << 4;

  // ---- cooperative panel fill: 2048 elems, 8 per thread, zero-masked ----
  {
    const int t = threadIdx.x;
    #pragma unroll
    for (int i = 0; i < (GK * 16) / 256; ++i) {
      int idx = t + (i << 8);                   // 0..2047
      int k   = idx >> 4;
      int col = idx & 15;
      int gc  = col0 + col;
      bp[idx] = (gc < C) ? W[(size_t)k * C + gc] : 0.0f;
    }
  }
  __syncthreads();

  const int lane = threadIdx.x & 31;
  const int wave = threadIdx.x >> 5;
  const int row0 = (blockIdx.x * 8 + wave) << 4;
  if (row0 < M) {                               // wave-uniform: EXEC stays full
    const int lh = lane >> 4;                   // lane half: K pair {0,1}|{2,3}
    const int ll = lane & 15;

    const float* Ap = A + (size_t)(row0 + ll) * GK + (lh << 1);
    const float* Bp = bp + ll;                  // VGPR base; imm offsets below
    v8f acc = {};
    #pragma unroll
    for (int kk = 0; kk < GK / 4; ++kk) {
      v2f a = *(const v2f*)(Ap + (kk << 2));    // A frag: b64, imm offset
      const int kb = (kk << 2) + (lh << 1);
      v2f b;
      b.x = Bp[(kb)     << 4];                  // ds_load, imm offset kb*64B
      b.y = Bp[(kb + 1) << 4];
      acc = __builtin_amdgcn_wmma_f32_16x16x4_f32(
          false, a, false, b, (short)0, acc, false, false);
    }

    // ---- epilogue: write h and agg = h*dinv^2 + bias ----
    const int bcol = col0 + ll;
    if (bcol < C) {
      const int rbase = row0 + (lh << 3);       // D rows = vgpr j + 8*lanehalf
      const float bb = bias[bcol];
      #pragma unroll
      for (int j = 0; j < 8; ++j) {
        float v  = acc[j];
        float di = dinv[rbase + j];
        size_t idx = (size_t)(rbase + j) * C + bcol;
        h[idx]   = v;
        agg[idx] = v * di * di + bb;
      }
    }
  }
}

// ------------------------------ edge scatter -------------------------------
// agg[dst,:] += h[src,:] * dinv[src]*dinv[dst]   (one wave32 per edge)
__global__ void __launch_bounds__(256)
k_scatter(const float* __restrict__ h, const int* __restrict__ src,
          const int* __restrict__ dst, const float* __restrict__ dinv,
          float* __restrict__ agg, int E, int C) {
  int e    = (int)((blockIdx.x * (unsigned)blockDim.x + threadIdx.x) >> 5);
  int lane = threadIdx.x & 31;
  if (e >= E) return;
  int s = src[e], d = dst[e];
  float nrm = dinv[s] * dinv[d];
  if (C == 128) {                               // fast path: b128 load + 4 atomics
    const float4* hp = (const float4*)(h + (size_t)s * 128) + lane;
    float*        ap = agg + (size_t)d * 128 + (lane << 2);
    float4 v = *hp;
    atomicAdd(ap + 0, v.x * nrm);
    atomicAdd(ap + 1, v.y * nrm);
    atomicAdd(ap + 2, v.z * nrm);
    atomicAdd(ap + 3, v.w * nrm);
  } else {
    const float* hp = h   + (size_t)s * C;
    float*       ap = agg + (size_t)d * C;
    for (int f = lane; f < C; f += 32)
      atomicAdd(&ap[f], hp[f] * nrm);
  }
}

// ------------------------------ BatchNorm ----------------------------------
// stats[c] = sum, stats[128+c] = sumsq over rows (C fixed at 128).
__global__ void __launch_bounds__(256)
k_bn_stats(const float* __restrict__ x, float* __restrict__ stats, int Nn) {
  __shared__ float shs[256];
  __shared__ float shq[256];
  const int c    = threadIdx.x & 127;
  const int half = threadIdx.x >> 7;
  float s = 0.0f, sq = 0.0f;
  for (int r = blockIdx.x * 2 + half; r < Nn; r += gridDim.x * 2) {
    float v = x[(size_t)r * 128 + c];
    s += v; sq += v * v;
  }
  shs[threadIdx.x] = s;
  shq[threadIdx.x] = sq;
  __syncthreads();
  if (half == 0) {
    atomicAdd(&stats[c],        shs[c] + shs[c + 128]);
    atomicAdd(&stats[128 + c],  shq[c] + shq[c + 128]);
  }
}

// ss[c] = gamma*rstd, ss[128+c] = beta - mu*gamma*rstd
__global__ void k_bn_finalize(const float* __restrict__ stats,
                              const float* __restrict__ gamma,
                              const float* __restrict__ beta,
                              float* __restrict__ ss, int Nn) {
  int c = threadIdx.x;  // 128 threads
  float invn = 1.0f / (float)Nn;
  float mu   = stats[c] * invn;
  float var  = stats[128 + c] * invn - mu * mu;   // population variance (ddof=0)
  float sc   = gamma[c] * rsqrtf(var + BN_EPS);
  ss[c]        = sc;
  ss[128 + c]  = beta[c] - mu * sc;
}

__global__ void k_bn_relu(const float* __restrict__ x, const float* __restrict__ ss,
                          float* __restrict__ y) {
  int i = blockIdx.x, c = threadIdx.x;  // 128 threads
  size_t idx = (size_t)i * 128 + c;
  float v = x[idx] * ss[c] + ss[128 + c];
  y[idx] = v > 0.0f ? v : 0.0f;
}

// ----------------------------- log_softmax ---------------------------------
// One wave32 per row; safe in-place (each element read once just before write).
__global__ void __launch_bounds__(256)
k_log_softmax(const float* __restrict__ x, float* __restrict__ out, int Nn, int C) {
  int row  = (int)((blockIdx.x * (unsigned)blockDim.x + threadIdx.x) >> 5);
  int lane = threadIdx.x & 31;
  if (row >= Nn) return;
  const float* xp = x + (size_t)row * C;
  float m = -INFINITY;
  for (int f = lane; f < C; f += 32) m = fmaxf(m, xp[f]);
  for (int o = 16; o > 0; o >>= 1)   m = fmaxf(m, __shfl_xor(m, o, 32));
  float s = 0.0f;
  for (int f = lane; f < C; f += 32) s += __expf(xp[f] - m);
  for (int o = 16; o > 0; o >>= 1)   s += __shfl_xor(s, o, 32);
  float lg = m + __logf(s);
  for (int f = lane; f < C; f += 32) out[(size_t)row * C + f] = xp[f] - lg;
}

// ------------------------------- driver ------------------------------------
extern "C" void kernel_launch(void* const* d_in, const int* in_sizes, int n_in,
                              void* d_out, int out_size, void* d_ws, size_t ws_size,
                              hipStream_t stream) {
  const float* x   = (const float*)d_in[0];
  const int*   ei  = (const int*)  d_in[1];
  const float* W1  = (const float*)d_in[2];
  const float* b1  = (const float*)d_in[3];
  const float* g1  = (const float*)d_in[4];
  const float* be1 = (const float*)d_in[5];
  const float* W2  = (const float*)d_in[6];
  const float* b2  = (const float*)d_in[7];
  const float* g2  = (const float*)d_in[8];
  const float* be2 = (const float*)d_in[9];
  const float* W3  = (const float*)d_in[10];
  const float* b3  = (const float*)d_in[11];
  float* out = (float*)d_out;

  const int Ch = in_sizes[3];          // 128 (== GK)
  const int Co = in_sizes[11];         // 40
  const int Nn = in_sizes[0] / Ch;     // 50000
  const int E  = in_sizes[1] / 2;      // 800000
  const int* src = ei;
  const int* dst = ei + E;

  // workspace layout (floats)
  float* ws   = (float*)d_ws;
  float* deg  = ws;                       // N
  float* dinv = deg  + Nn;                // N
  float* stat = dinv + Nn;                // 256
  float* ss   = stat + 256;               // 256
  float* bufA = ss   + 256;               // N*Ch
  float* bufB = bufA + (size_t)Nn * Ch;   // N*Ch
  float* bufC = bufB + (size_t)Nn * Ch;   // N*Ch

  const int mtiles = (Nn + 15) / 16;
  dim3 ggh((mtiles + 7) / 8, Ch / 16);          // hidden GEMMs: 8 col tiles
  dim3 ggo((mtiles + 7) / 8, (Co + 15) / 16);   // output GEMM:  3 col tiles
  const int sgrid = (E + 7) / 8;                // scatter: 8 edges (waves)/block

  // ---- normalization coefficients ----
  k_zero  <<<(Nn + 255) / 256, 256, 0, stream>>>(deg, Nn);
  k_degree<<<(E  + 255) / 256, 256, 0, stream>>>(dst, deg, E);
  k_dinv  <<<(Nn + 255) / 256, 256, 0, stream>>>(deg, dinv, Nn);

  // ---- layer 1: GCNConv (h1->bufA, agg1 init->bufB) + BN + ReLU ----
  k_gemm_wmma  <<<ggh, 256, 0, stream>>>(x, W1, dinv, b1, bufA, bufB, Nn, Ch);
  k_scatter    <<<sgrid, 256, 0, stream>>>(bufA, src, dst, dinv, bufB, E, Ch);
  k_zero       <<<1, 256, 0, stream>>>(stat, 256);
  k_bn_stats   <<<512, 256, 0, stream>>>(bufB, stat, Nn);
  k_bn_finalize<<<1, 128, 0, stream>>>(stat, g1, be1, ss, Nn);
  k_bn_relu    <<<Nn, 128, 0, stream>>>(bufB, ss, bufA);

  // ---- layer 2: GCNConv (h2->bufB, agg2 init->bufC) + BN + ReLU ----
  k_gemm_wmma  <<<ggh, 256, 0, stream>>>(bufA, W2, dinv, b2, bufB, bufC, Nn, Ch);
  k_scatter    <<<sgrid, 256, 0, stream>>>(bufB, src, dst, dinv, bufC, E, Ch);
  k_zero       <<<1, 256, 0, stream>>>(stat, 256);
  k_bn_stats   <<<512, 256, 0, stream>>>(bufC, stat, Nn);
  k_bn_finalize<<<1, 128, 0, stream>>>(stat, g2, be2, ss, Nn);
  k_bn_relu    <<<Nn, 128, 0, stream>>>(bufC, ss, bufA);

  // ---- layer 3: GCNConv (h3->bufB, agg3 init->d_out) + log_softmax ----
  k_gemm_wmma  <<<ggo, 256, 0, stream>>>(bufA, W3, dinv, b3, bufB, out, Nn, Co);
  k_scatter    <<<sgrid, 256, 0, stream>>>(bufB, src, dst, dinv, out, E, Co);
  k_log_softmax<<<(Nn + 7) / 8, 256, 0, stream>>>(out, out, Nn, Co);
}